// LatticeGAT_28527172780200
// MI455X (gfx1250) — compile-verified
//
#include <hip/hip_runtime.h>
#include <hip/hip_bf16.h>
#include <math.h>

#define IN_CH 128
#define HID 64
#define SLOPE_ATT 0.2f
#define SLOPE_ACT 0.01f

typedef __attribute__((ext_vector_type(2))) float v2f;
typedef __attribute__((ext_vector_type(8))) float v8f;

// ---------------------------------------------------------------------------
// H[n, HID] = X[n, K] @ W[K, HID]   via V_WMMA_F32_16X16X4_F32 (full f32 math)
// grid.x = ceil(n/32), block = 256 (8 waves): wave covers one 16x16 tile,
// 2 waves in M x 4 waves in N => block tile is 32 rows x 64 cols.
// ---------------------------------------------------------------------------
__global__ __launch_bounds__(256)
void gat_linear_wmma(const float* __restrict__ X, const float* __restrict__ W,
                     float* __restrict__ H, int n, int K) {
  const int lane = threadIdx.x & 31;
  const int wave = threadIdx.x >> 5;
  const int row0 = blockIdx.x * 32 + (wave >> 2) * 16;  // M tile base
  const int col0 = (wave & 3) * 16;                     // N tile base
  // A fragment addressing: lanes 0-15 -> M=lane, K pair {0,1};
  //                        lanes 16-31 -> M=lane-16, K pair {2,3}
  const int mrow = row0 + (lane & 15);
  const int arow = (mrow < n) ? mrow : (n - 1);         // clamp: EXEC must stay all-1s
  const int kk   = (lane >> 4) * 2;
  const int ncol = col0 + (lane & 15);                  // B / C column for this lane

  const float* xr = X + (size_t)arow * K;
  v8f acc = {};
  for (int k0 = 0; k0 < K; k0 += 4) {
    v2f a, b;
    a.x = xr[k0 + kk];
    a.y = xr[k0 + kk + 1];
    b.x = W[(size_t)(k0 + kk) * HID + ncol];
    b.y = W[(size_t)(k0 + kk + 1) * HID + ncol];
    // 8 args: (neg_a, A, neg_b, B, c_mod, C, reuse_a, reuse_b)
    acc = __builtin_amdgcn_wmma_f32_16x16x4_f32(false, a, false, b,
                                                (short)0, acc, false, false);
  }
  // C/D layout: VGPR r -> M = r + 8*(lane>=16), N = lane&15
  const int rbase = row0 + (lane >> 4) * 8;
#pragma unroll
  for (int r = 0; r < 8; ++r) {
    int row = rbase + r;
    if (row < n) H[(size_t)row * HID + ncol] = acc[r];
  }
}

// ---------------------------------------------------------------------------
// Per-node attention logit halves: al_s[i] = h_i . a_src ; al_d[i] = h_i . a_dst
// ---------------------------------------------------------------------------
__global__ void gat_attn(const float* __restrict__ H, const float* __restrict__ a_s,
                         const float* __restrict__ a_d, float* __restrict__ als,
                         float* __restrict__ ald, int n) {
  int i = blockIdx.x * blockDim.x + threadIdx.x;
  if (i >= n) return;
  const float* h = H + (size_t)i * HID;
  float s = 0.f, d = 0.f;
#pragma unroll
  for (int c = 0; c < HID; ++c) { float v = h[c]; s += v * a_s[c]; d += v * a_d[c]; }
  als[i] = s;
  ald[i] = d;
}

// ---------------------------------------------------------------------------
// Reset per-layer accumulators (agg=0, m=-inf, denom=0)
// ---------------------------------------------------------------------------
__global__ void gat_init(float* __restrict__ agg, float* __restrict__ m,
                         float* __restrict__ den, int n) {
  int i = blockIdx.x * blockDim.x + threadIdx.x;
  if (i < n * HID) agg[i] = 0.f;
  if (i < n) { m[i] = -INFINITY; den[i] = 0.f; }
}

// Order-preserving float max via integer atomics (always available, and
// correct for mixed signs when initialized to -inf).
__device__ __forceinline__ void atomicMaxF32(float* addr, float v) {
  if (v >= 0.f) atomicMax((int*)addr, __float_as_int(v));
  else          atomicMin((unsigned int*)addr, __float_as_uint(v));
}

// Shared per-edge logit recompute (cheaper than an E-sized exp buffer: the
// al_s/al_d tables are 200KB each and L2-resident).
__device__ __forceinline__ float edge_logit(const long long* __restrict__ src,
                                            const long long* __restrict__ dst,
                                            const float* __restrict__ als,
                                            const float* __restrict__ ald,
                                            int e, int nE, int* s_out, int* d_out) {
  int s, d;
  if (e < nE) { s = (int)src[e]; d = (int)dst[e]; }
  else        { s = e - nE; d = s; }                 // implicit self-loops
  float v = als[s] + ald[d];
  v = v > 0.f ? v : SLOPE_ATT * v;                   // leaky_relu(0.2)
  *s_out = s; *d_out = d;
  return v;
}

__global__ void gat_edge_max(const long long* __restrict__ src,
                             const long long* __restrict__ dst,
                             const float* __restrict__ als, const float* __restrict__ ald,
                             float* __restrict__ m, int nE, int n) {
  int e = blockIdx.x * blockDim.x + threadIdx.x;
  if (e >= nE + n) return;
  int s, d;
  float v = edge_logit(src, dst, als, ald, e, nE, &s, &d);
  atomicMaxF32(&m[d], v);
}

__global__ void gat_edge_expsum(const long long* __restrict__ src,
                                const long long* __restrict__ dst,
                                const float* __restrict__ als, const float* __restrict__ ald,
                                const float* __restrict__ m, float* __restrict__ den,
                                int nE, int n) {
  int e = blockIdx.x * blockDim.x + threadIdx.x;
  if (e >= nE + n) return;
  int s, d;
  float v = edge_logit(src, dst, als, ald, e, nE, &s, &d);
  atomicAdd(&den[d], __expf(v - m[d]));
}

// One wave (32 lanes) per edge; each lane handles channels {lane, lane+32}.
// h[src] rows are contiguous -> coalesced 128B loads; atomics land in L2.
__global__ __launch_bounds__(256)
void gat_edge_agg(const long long* __restrict__ src, const long long* __restrict__ dst,
                  const float* __restrict__ als, const float* __restrict__ ald,
                  const float* __restrict__ m, const float* __restrict__ den,
                  const float* __restrict__ H, float* __restrict__ agg,
                  int nE, int n) {
  long long t = (long long)blockIdx.x * blockDim.x + threadIdx.x;
  int e    = (int)(t >> 5);
  int lane = (int)(t & 31);
  if (e >= nE + n) return;
  int s, d;
  float v = edge_logit(src, dst, als, ald, e, nE, &s, &d);
  float alpha = __expf(v - m[d]) / den[d];
  const float* hs = H + (size_t)s * HID;
  float* od = agg + (size_t)d * HID;
  atomicAdd(&od[lane],      alpha * hs[lane]);
  atomicAdd(&od[lane + 32], alpha * hs[lane + 32]);
}

__global__ void gat_bias_act(float* __restrict__ agg, const float* __restrict__ b,
                             int total) {
  int i = blockIdx.x * blockDim.x + threadIdx.x;
  if (i >= total) return;
  float v = agg[i] + b[i & (HID - 1)];
  agg[i] = v > 0.f ? v : SLOPE_ACT * v;              // leaky_relu(0.01)
}

__global__ void gat_head(const float* __restrict__ H, const float* __restrict__ Wo,
                         const float* __restrict__ bo, float* __restrict__ out, int n) {
  int i = blockIdx.x * blockDim.x + threadIdx.x;
  if (i >= n) return;
  const float* h = H + (size_t)i * HID;
  float acc = bo[0];
#pragma unroll
  for (int c = 0; c < HID; ++c) acc += h[c] * Wo[c];
  out[i] = acc;
}

// ---------------------------------------------------------------------------
extern "C" void kernel_launch(void* const* d_in, const int* in_sizes, int n_in,
                              void* d_out, int out_size, void* d_ws, size_t ws_size,
                              hipStream_t stream) {
  const float*     x   = (const float*)d_in[0];
  const long long* ei  = (const long long*)d_in[1];   // int64 edge_index [2, E]
  const int N = in_sizes[0] / IN_CH;
  const int E = in_sizes[1] / 2;
  const long long* src = ei;
  const long long* dst = ei + E;

  const float* Wl[3]  = {(const float*)d_in[2],  (const float*)d_in[6],  (const float*)d_in[10]};
  const float* asl[3] = {(const float*)d_in[3],  (const float*)d_in[7],  (const float*)d_in[11]};
  const float* adl[3] = {(const float*)d_in[4],  (const float*)d_in[8],  (const float*)d_in[12]};
  const float* bl[3]  = {(const float*)d_in[5],  (const float*)d_in[9],  (const float*)d_in[13]};
  const float* Wo = (const float*)d_in[14];
  const float* bo = (const float*)d_in[15];
  float* out = (float*)d_out;

  // Workspace: two N x HID buffers + 4 N-sized scalar tables (~26.5 MB total)
  float* bufL = (float*)d_ws;                 // linear output h (per layer)
  float* bufH = bufL + (size_t)N * HID;       // layer input / aggregation output
  float* als  = bufH + (size_t)N * HID;
  float* ald  = als + N;
  float* mseg = ald + N;
  float* den  = mseg + N;

  const int T = 256;
  const int nTot = E + N;                     // edges incl. self-loops
  const int gN    = (N + T - 1) / T;
  const int gNH   = (N * HID + T - 1) / T;
  const int gEdge = (nTot + T - 1) / T;
  const long long aggThreads = (long long)nTot * 32;
  const unsigned gAgg = (unsigned)((aggThreads + T - 1) / T);

  const float* cur = x;
  int K = IN_CH;
  for (int l = 0; l < 3; ++l) {
    gat_linear_wmma<<<(N + 31) / 32, T, 0, stream>>>(cur, Wl[l], bufL, N, K);
    gat_attn<<<gN, T, 0, stream>>>(bufL, asl[l], adl[l], als, ald, N);
    gat_init<<<gNH, T, 0, stream>>>(bufH, mseg, den, N);     // bufH consumed by linear above
    gat_edge_max<<<gEdge, T, 0, stream>>>(src, dst, als, ald, mseg, E, N);
    gat_edge_expsum<<<gEdge, T, 0, stream>>>(src, dst, als, ald, mseg, den, E, N);
    gat_edge_agg<<<gAgg, T, 0, stream>>>(src, dst, als, ald, mseg, den, bufL, bufH, E, N);
    gat_bias_act<<<gNH, T, 0, stream>>>(bufH, bl[l], N * HID);
    cur = bufH;
    K = HID;
  }
  gat_head<<<gN, T, 0, stream>>>(bufH, Wo, bo, out, N);
}